// GCN_9079560864488
// MI455X (gfx1250) — compile-verified
//
#include <hip/hip_runtime.h>

// GCN forward for MI455X (gfx1250, wave32).
// Dense GEMMs use V_WMMA_F32_16X16X4_F32 (exact fp32 matrix path).
// Edge aggregation / pooling use global_atomic_add_f32; feature matrix (41MB)
// is L2-resident (192MB L2) so random gather/scatter stays on-chip.

typedef __attribute__((ext_vector_type(2))) float v2f;
typedef __attribute__((ext_vector_type(8))) float v8f;

#define NN     40000
#define EE     640000
#define INC    128
#define HID    256
#define OUTC   10
#define GG     64

// ---------------------------------------------------------------- utilities
__global__ __launch_bounds__(256) void fill_kernel(float* __restrict__ p, float v, int n) {
    int i = blockIdx.x * 256 + threadIdx.x;
    if (i < n) p[i] = v;
}

__global__ __launch_bounds__(256) void deg_count_kernel(const int* __restrict__ dst,
                                                        float* __restrict__ deg, int e) {
    int i = blockIdx.x * 256 + threadIdx.x;
    if (i < e) atomicAdd(&deg[dst[i]], 1.0f);
}

__global__ __launch_bounds__(256) void dis_kernel(float* __restrict__ degdis, int n) {
    int i = blockIdx.x * 256 + threadIdx.x;
    if (i < n) degdis[i] = rsqrtf(fmaxf(degdis[i], 1.0f));   // deg includes self-loop
}

// ---------------------------------------------------------------- WMMA GEMM
// Y[nrows x 256] = X[nrows x K] @ W[K x 256], fp32, V_WMMA_F32_16X16X4_F32.
// Wave computes a 16x64 strip (4 accumulators, A reused). Block = 8 waves ->
// 64 rows x 128 cols. Grid: (nrows/64, 256/128).
template<int K>
__global__ __launch_bounds__(256) void gemm_wmma_kernel(const float* __restrict__ X,
                                                        const float* __restrict__ W,
                                                        float* __restrict__ Y) {
    const int lane    = threadIdx.x & 31;
    const int wave    = threadIdx.x >> 5;                 // 0..7
    const int rowBase = blockIdx.x * 64 + (wave >> 1) * 16;
    const int colBase = blockIdx.y * 128 + (wave & 1) * 64;

    const int m    = lane & 15;                           // A row / B-&-C col within tile
    const int koff = (lane >> 4) << 1;                    // 0 or 2 (K half per lane group)

    v8f acc0 = {}, acc1 = {}, acc2 = {}, acc3 = {};

    const float* xrow  = X + (size_t)(rowBase + m) * K + koff;
    const float* wbase = W + (size_t)koff * HID + colBase + m;

    for (int k = 0; k < K; k += 4) {
        v2f a = *(const v2f*)(xrow + k);                  // A[m][k+koff .. +1]
        const float* wp = wbase + (size_t)k * HID;
        v2f b0 = { wp[ 0], wp[ 0 + HID] };                // B[k+koff][col], B[k+koff+1][col]
        v2f b1 = { wp[16], wp[16 + HID] };
        v2f b2 = { wp[32], wp[32 + HID] };
        v2f b3 = { wp[48], wp[48 + HID] };
        acc0 = __builtin_amdgcn_wmma_f32_16x16x4_f32(false, a, false, b0, (short)0, acc0, false, false);
        acc1 = __builtin_amdgcn_wmma_f32_16x16x4_f32(false, a, false, b1, (short)0, acc1, false, false);
        acc2 = __builtin_amdgcn_wmma_f32_16x16x4_f32(false, a, false, b2, (short)0, acc2, false, false);
        acc3 = __builtin_amdgcn_wmma_f32_16x16x4_f32(false, a, false, b3, (short)0, acc3, false, false);
    }

    // C/D layout: acc[r] -> row = r + 8*(lane>=16), col = lane&15
    float* yp = Y + (size_t)(rowBase + ((lane >> 4) << 3)) * HID + colBase + m;
#pragma unroll
    for (int r = 0; r < 8; ++r) {
        yp[(size_t)r * HID +  0] = acc0[r];
        yp[(size_t)r * HID + 16] = acc1[r];
        yp[(size_t)r * HID + 32] = acc2[r];
        yp[(size_t)r * HID + 48] = acc3[r];
    }
}

// ---------------------------------------------------------------- aggregation
// One wave per (edge or self-loop): OUT[dst] += XW[src] * dis[src]*dis[dst].
__global__ __launch_bounds__(256) void aggregate_kernel(const float* __restrict__ XW,
                                                        const int* __restrict__ srcI,
                                                        const int* __restrict__ dstI,
                                                        const float* __restrict__ dis,
                                                        float* __restrict__ OUT,
                                                        int e, int n) {
    int edge = blockIdx.x * 8 + (threadIdx.x >> 5);
    int lane = threadIdx.x & 31;
    if (edge >= e + n) return;
    int s, d;
    if (edge < e) { s = srcI[edge]; d = dstI[edge]; }
    else          { s = d = edge - e; }                   // self-loop
    float norm = dis[s] * dis[d];
    const float4* sp = (const float4*)(XW + (size_t)s * HID);
    float* dp = OUT + (size_t)d * HID;
#pragma unroll
    for (int j = 0; j < 2; ++j) {
        int idx = lane + j * 32;                          // float4 index, 256 floats total
        float4 v = sp[idx];
        atomicAdd(dp + idx * 4 + 0, v.x * norm);
        atomicAdd(dp + idx * 4 + 1, v.y * norm);
        atomicAdd(dp + idx * 4 + 2, v.z * norm);
        atomicAdd(dp + idx * 4 + 3, v.w * norm);
    }
}

__global__ __launch_bounds__(256) void bias_relu_kernel(float* __restrict__ h,
                                                        const float* __restrict__ b, int total) {
    int i = blockIdx.x * 256 + threadIdx.x;
    if (i < total) h[i] = fmaxf(h[i] + b[i & (HID - 1)], 0.0f);
}

// ---------------------------------------------------------------- pooling
__global__ __launch_bounds__(256) void pool_kernel(const float* __restrict__ h,
                                                   const int* __restrict__ batch,
                                                   float* __restrict__ pooled,
                                                   float* __restrict__ cnts, int n) {
    int node = blockIdx.x * 8 + (threadIdx.x >> 5);
    int lane = threadIdx.x & 31;
    if (node >= n) return;
    int g = batch[node];
    const float* hp = h + (size_t)node * HID;
    float* pp = pooled + (size_t)g * HID;
#pragma unroll
    for (int j = 0; j < 8; ++j) {
        int idx = lane + j * 32;
        atomicAdd(pp + idx, hp[idx]);
    }
    if (lane == 0) atomicAdd(cnts + g, 1.0f);
}

// out[g][o] = (pooled_sum[g]/cnt[g]) @ Wlin + blin
__global__ __launch_bounds__(128) void final_linear_kernel(const float* __restrict__ pooled,
                                                           const float* __restrict__ cnts,
                                                           const float* __restrict__ Wlin,
                                                           const float* __restrict__ blin,
                                                           float* __restrict__ out) {
    int t = blockIdx.x * 128 + threadIdx.x;
    if (t >= GG * OUTC) return;
    int g = t / OUTC, o = t % OUTC;
    float inv = 1.0f / fmaxf(cnts[g], 1.0f);
    const float* pr = pooled + (size_t)g * HID;
    float acc = 0.0f;
    for (int j = 0; j < HID; ++j) acc += pr[j] * Wlin[j * OUTC + o];
    out[t] = acc * inv + blin[o];
}

// ---------------------------------------------------------------- launcher
extern "C" void kernel_launch(void* const* d_in, const int* in_sizes, int n_in,
                              void* d_out, int out_size, void* d_ws, size_t ws_size,
                              hipStream_t stream) {
    (void)in_sizes; (void)n_in; (void)out_size; (void)ws_size;
    const float* x    = (const float*)d_in[0];
    const int*   ei   = (const int*)d_in[1];      // [2,E] flat: src rows then dst rows
    const int*   batch= (const int*)d_in[2];
    const float* W1   = (const float*)d_in[3];
    const float* b1   = (const float*)d_in[4];
    const float* W2   = (const float*)d_in[5];
    const float* b2   = (const float*)d_in[6];
    const float* Wlin = (const float*)d_in[7];
    const float* blin = (const float*)d_in[8];
    float* out = (float*)d_out;

    const int* srcI = ei;
    const int* dstI = ei + EE;

    // workspace layout (all 256B aligned by construction)
    char* ws = (char*)d_ws;
    size_t featBytes = (size_t)NN * HID * sizeof(float);   // 40,960,000
    float* buf0   = (float*)(ws);                          // XW scratch
    float* buf1   = (float*)(ws + featBytes);              // AGG / H scratch
    float* dis    = (float*)(ws + 2 * featBytes);          // N floats (deg -> dis in place)
    float* pooled = (float*)(ws + 2 * featBytes + ((size_t)NN * 4 + 255) / 256 * 256);
    float* cnts   = pooled + (size_t)GG * HID;

    const int featN = NN * HID;                 // 10,240,000
    dim3 blk(256);
    dim3 gFeat((featN + 255) / 256);
    dim3 gNode((NN + 255) / 256);
    dim3 gEdgeCnt((EE + 255) / 256);
    dim3 gGemm(NN / 64, HID / 128);             // 625 x 2
    dim3 gAgg((EE + NN + 7) / 8);               // wave per edge, 8 waves/block
    dim3 gPool((NN + 7) / 8);

    // normalization: deg (with self loop) -> dis = rsqrt(deg)
    fill_kernel<<<gNode, blk, 0, stream>>>(dis, 1.0f, NN);
    deg_count_kernel<<<gEdgeCnt, blk, 0, stream>>>(dstI, dis, EE);
    dis_kernel<<<gNode, blk, 0, stream>>>(dis, NN);

    // ---- layer 1: h1 = relu(Agg(x@W1) + b1)
    gemm_wmma_kernel<INC><<<gGemm, blk, 0, stream>>>(x, W1, buf0);
    fill_kernel<<<gFeat, blk, 0, stream>>>(buf1, 0.0f, featN);
    aggregate_kernel<<<gAgg, blk, 0, stream>>>(buf0, srcI, dstI, dis, buf1, EE, NN);
    bias_relu_kernel<<<gFeat, blk, 0, stream>>>(buf1, b1, featN);

    // ---- layer 2: h2 = relu(Agg(h1@W2) + b2)
    gemm_wmma_kernel<HID><<<gGemm, blk, 0, stream>>>(buf1, W2, buf0);
    fill_kernel<<<gFeat, blk, 0, stream>>>(buf1, 0.0f, featN);   // buf1 reused as AGG2
    aggregate_kernel<<<gAgg, blk, 0, stream>>>(buf0, srcI, dstI, dis, buf1, EE, NN);
    bias_relu_kernel<<<gFeat, blk, 0, stream>>>(buf1, b2, featN);

    // ---- mean pool + final linear
    fill_kernel<<<dim3((GG * HID + GG + 255) / 256), blk, 0, stream>>>(pooled, 0.0f, GG * HID + GG);
    pool_kernel<<<gPool, blk, 0, stream>>>(buf1, batch, pooled, cnts, NN);
    final_linear_kernel<<<dim3((GG * OUTC + 127) / 128), dim3(128), 0, stream>>>(
        pooled, cnts, Wlin, blin, out);
}